// GRAND_79809082294826
// MI455X (gfx1250) — compile-verified
//
#include <hip/hip_runtime.h>
#include <hip/hip_bf16.h>
#include <math.h>

typedef __attribute__((ext_vector_type(2))) float v2f;
typedef __attribute__((ext_vector_type(8))) float v8f;

#define IN_DIM   128
#define HID_DIM  256
#define N_CLASS  40

static __device__ __forceinline__ v8f wmma_f32(v2f a, v2f b, v8f c) {
  // V_WMMA_F32_16X16X4_F32: D = A(16x4) * B(4x16) + C(16x16), all f32
  return __builtin_amdgcn_wmma_f32_16x16x4_f32(false, a, false, b, (short)0, c, false, false);
}

// ---------------- utility kernels ----------------

__global__ void zero_i32_k(int* __restrict__ p, int n) {
  int i = blockIdx.x * blockDim.x + threadIdx.x;
  if (i < n) p[i] = 0;
}

__global__ void deg_k(const int* __restrict__ dst, int* __restrict__ degi, int E) {
  int e = blockIdx.x * blockDim.x + threadIdx.x;
  if (e < E) atomicAdd(&degi[dst[e]], 1);
}

// single-workgroup exclusive scan over N=50000 ints (chunked Hillis-Steele)
__global__ void scan_k(const int* __restrict__ degi, int* __restrict__ rowptr, int n) {
  __shared__ int tmp[1024];
  __shared__ int carry;
  int tid = threadIdx.x;
  if (tid == 0) carry = 0;
  __syncthreads();
  for (int base = 0; base < n; base += 1024) {
    int v = (base + tid < n) ? degi[base + tid] : 0;
    tmp[tid] = v;
    __syncthreads();
    for (int off = 1; off < 1024; off <<= 1) {
      int t = (tid >= off) ? tmp[tid - off] : 0;
      __syncthreads();
      tmp[tid] += t;
      __syncthreads();
    }
    if (base + tid < n) rowptr[base + tid] = carry + tmp[tid] - v;  // exclusive
    __syncthreads();
    if (tid == 0) carry += tmp[1023];
    __syncthreads();
  }
  if (threadIdx.x == 0) rowptr[n] = carry;
}

__global__ void norm_k(const int* __restrict__ degi, float* __restrict__ nrm, int n) {
  int i = blockIdx.x * blockDim.x + threadIdx.x;
  if (i < n) nrm[i] = rsqrtf(fmaxf((float)degi[i], 1.0f));
}

// bucket edges by destination: es[p]=src, ew[p]=norm[src]*norm[dst]
__global__ void bucket_k(const int* __restrict__ src, const int* __restrict__ dst,
                         const int* __restrict__ rowptr, int* __restrict__ fill,
                         const float* __restrict__ nrm,
                         int* __restrict__ es, float* __restrict__ ew, int E) {
  int e = blockIdx.x * blockDim.x + threadIdx.x;
  if (e >= E) return;
  int d = dst[e], s = src[e];
  int p = rowptr[d] + atomicAdd(&fill[d], 1);
  es[p] = s;
  ew[p] = nrm[s] * nrm[d];
}

__global__ void copy4_k(const float4* __restrict__ f, float4* __restrict__ y, int n4) {
  int i = blockIdx.x * blockDim.x + threadIdx.x;
  if (i < n4) y[i] = f[i];
}

// one wave per node: xn[n] = sum_{j in CSR row n} ew[j] * x[es[j]]; y += xn (fused)
__global__ __launch_bounds__(256)
void prop_k(const float* __restrict__ x, const int* __restrict__ rowptr,
            const int* __restrict__ es, const float* __restrict__ ew,
            float* __restrict__ xn, float* __restrict__ y, int n) {
  int wave = blockIdx.x * (blockDim.x >> 5) + (threadIdx.x >> 5);
  if (wave >= n) return;
  int lane = threadIdx.x & 31;
  int beg = rowptr[wave], end = rowptr[wave + 1];
  float4 acc = make_float4(0.f, 0.f, 0.f, 0.f);
  for (int j = beg; j < end; ++j) {
    int s = es[j];
    float wt = ew[j];
    const float4 v = *(const float4*)(x + (size_t)s * IN_DIM + lane * 4);
    acc.x = fmaf(v.x, wt, acc.x);
    acc.y = fmaf(v.y, wt, acc.y);
    acc.z = fmaf(v.z, wt, acc.z);
    acc.w = fmaf(v.w, wt, acc.w);
  }
  size_t o = (size_t)wave * IN_DIM + lane * 4;
  *(float4*)(xn + o) = acc;
  float4 yv = *(const float4*)(y + o);
  yv.x += acc.x; yv.y += acc.y; yv.z += acc.z; yv.w += acc.w;
  *(float4*)(y + o) = yv;
}

// ---------------- WMMA GEMMs ----------------
// H = relu((Y/4) @ W1 + b1) ; one wave -> 16x64 strip (4 accumulators)
__global__ __launch_bounds__(256)
void gemm1_k(const float* __restrict__ Y, const float* __restrict__ W1,
             const float* __restrict__ B1, float* __restrict__ H, int mtiles) {
  int wave = blockIdx.x * (blockDim.x >> 5) + (threadIdx.x >> 5);
  int mt = wave >> 2;       // 4 N-groups of 64 cols (HID_DIM=256)
  int ng = wave & 3;
  if (mt >= mtiles) return;
  int lane = threadIdx.x & 31;
  int half = lane >> 4;
  int lrow = lane & 15;
  const float* arow  = Y  + (size_t)(mt * 16 + lrow) * IN_DIM;
  const float* bbase = W1 + ng * 64 + lrow;
  v8f acc[4] = {};
#pragma unroll 8
  for (int k = 0; k < IN_DIM; k += 4) {
    int kk = k + 2 * half;
    v2f a;
    a.x = arow[kk];
    a.y = arow[kk + 1];
    const float* b0 = bbase + (size_t)kk * HID_DIM;
    const float* b1 = b0 + HID_DIM;
#pragma unroll
    for (int t = 0; t < 4; ++t) {
      v2f b;
      b.x = b0[t * 16];
      b.y = b1[t * 16];
      acc[t] = wmma_f32(a, b, acc[t]);
    }
  }
#pragma unroll
  for (int t = 0; t < 4; ++t) {
    int col = ng * 64 + t * 16 + lrow;
    float bias = B1[col];
#pragma unroll
    for (int v = 0; v < 8; ++v) {
      float r = fmaxf(acc[t][v] * 0.25f + bias, 0.0f);  // /4 + bias + relu fused
      H[(size_t)(mt * 16 + v + 8 * half) * HID_DIM + col] = r;
    }
  }
}

// L = H @ W2 + b2 ; one wave -> 16x48 strip (40 valid cols), guarded pad
__global__ __launch_bounds__(256)
void gemm2_k(const float* __restrict__ H, const float* __restrict__ W2,
             const float* __restrict__ B2, float* __restrict__ L, int mtiles) {
  int wave = blockIdx.x * (blockDim.x >> 5) + (threadIdx.x >> 5);
  if (wave >= mtiles) return;
  int lane = threadIdx.x & 31;
  int half = lane >> 4;
  int lrow = lane & 15;
  const float* arow = H + (size_t)(wave * 16 + lrow) * HID_DIM;
  v8f acc[3] = {};
#pragma unroll 8
  for (int k = 0; k < HID_DIM; k += 4) {
    int kk = k + 2 * half;
    v2f a;
    a.x = arow[kk];
    a.y = arow[kk + 1];
    const float* b0 = W2 + (size_t)kk * N_CLASS;
    const float* b1 = b0 + N_CLASS;
#pragma unroll
    for (int t = 0; t < 3; ++t) {
      int col = t * 16 + lrow;
      v2f b;
      b.x = (col < N_CLASS) ? b0[col] : 0.0f;
      b.y = (col < N_CLASS) ? b1[col] : 0.0f;
      acc[t] = wmma_f32(a, b, acc[t]);
    }
  }
#pragma unroll
  for (int t = 0; t < 3; ++t) {
    int col = t * 16 + lrow;
    if (col < N_CLASS) {
      float bias = B2[col];
#pragma unroll
      for (int v = 0; v < 8; ++v)
        L[(size_t)(wave * 16 + v + 8 * half) * N_CLASS + col] = acc[t][v] + bias;
    }
  }
}

__global__ void lsm_k(const float* __restrict__ L, float* __restrict__ out, int n) {
  int i = blockIdx.x * blockDim.x + threadIdx.x;
  if (i >= n) return;
  const float* r = L + (size_t)i * N_CLASS;
  float m = -3.402823466e38f;
  for (int c = 0; c < N_CLASS; ++c) m = fmaxf(m, r[c]);
  float s = 0.0f;
  for (int c = 0; c < N_CLASS; ++c) s += expf(r[c] - m);
  float ls = logf(s);
  float* o = out + (size_t)i * N_CLASS;
  for (int c = 0; c < N_CLASS; ++c) o[c] = r[c] - m - ls;
}

// ---------------- launch ----------------

extern "C" void kernel_launch(void* const* d_in, const int* in_sizes, int n_in,
                              void* d_out, int out_size, void* d_ws, size_t ws_size,
                              hipStream_t stream) {
  const float* feat = (const float*)d_in[0];
  const int*   src  = (const int*)d_in[1];
  const int*   dst  = (const int*)d_in[2];
  const float* W1   = (const float*)d_in[3];
  const float* b1   = (const float*)d_in[4];
  const float* W2   = (const float*)d_in[5];
  const float* b2   = (const float*)d_in[6];

  const int N = in_sizes[0] / IN_DIM;
  const int E = in_sizes[1];

  char* base = (char*)d_ws;
  size_t off = 0;
  auto carve = [&](size_t bytes) -> char* {
    char* p = base + off;
    off += (bytes + 255) & ~(size_t)255;
    return p;
  };
  int*   degi   = (int*)  carve((size_t)N * 4);
  int*   fill   = (int*)  carve((size_t)N * 4);
  int*   rowptr = (int*)  carve(((size_t)N + 1) * 4);
  float* nrm    = (float*)carve((size_t)N * 4);
  int*   es     = (int*)  carve((size_t)E * 4);
  float* ew     = (float*)carve((size_t)E * 4);
  float* xA     = (float*)carve((size_t)N * IN_DIM * 4);
  float* xB     = (float*)carve((size_t)N * IN_DIM * 4);
  float* yy     = (float*)carve((size_t)N * IN_DIM * 4);
  float* hh     = (float*)carve((size_t)N * HID_DIM * 4);
  float* lg     = (float*)carve((size_t)N * N_CLASS * 4);
  (void)ws_size; (void)n_in; (void)out_size;

  const int tb = 256;

  // CSR build (re-done every call: deterministic)
  zero_i32_k<<<(N + tb - 1) / tb, tb, 0, stream>>>(degi, N);
  zero_i32_k<<<(N + tb - 1) / tb, tb, 0, stream>>>(fill, N);
  deg_k<<<(E + tb - 1) / tb, tb, 0, stream>>>(dst, degi, E);
  scan_k<<<1, 1024, 0, stream>>>(degi, rowptr, N);
  norm_k<<<(N + tb - 1) / tb, tb, 0, stream>>>(degi, nrm, N);
  bucket_k<<<(E + tb - 1) / tb, tb, 0, stream>>>(src, dst, rowptr, fill, nrm, es, ew, E);

  // y = features (DROPOUT=0 -> scale 1)
  int n4 = (N * IN_DIM) / 4;
  copy4_k<<<(n4 + tb - 1) / tb, tb, 0, stream>>>((const float4*)feat, (float4*)yy, n4);

  // 3 atomic-free propagation steps, y accumulation fused
  int pblocks = (N + 7) / 8;  // 8 waves/block, 1 wave/node
  prop_k<<<pblocks, tb, 0, stream>>>(feat, rowptr, es, ew, xA, yy, N);
  prop_k<<<pblocks, tb, 0, stream>>>(xA,   rowptr, es, ew, xB, yy, N);
  prop_k<<<pblocks, tb, 0, stream>>>(xB,   rowptr, es, ew, xA, yy, N);

  // MLP on WMMA f32 path
  int mtiles = N / 16;                       // 3125 (N=50000 divides exactly)
  int g1waves = mtiles * 4;                  // 4 N-groups of 64 cols
  gemm1_k<<<(g1waves + 7) / 8, tb, 0, stream>>>(yy, W1, b1, hh, mtiles);
  gemm2_k<<<(mtiles + 7) / 8, tb, 0, stream>>>(hh, W2, b2, lg, mtiles);

  lsm_k<<<(N + tb - 1) / tb, tb, 0, stream>>>(lg, (float*)d_out, N);
}